// Qwen2DecoderLayer_56401510531128
// MI455X (gfx1250) — compile-verified
//
#include <hip/hip_runtime.h>
#include <math.h>

#define HID 3584
#define NH 28
#define NKV 4
#define HD 128
#define INTER 18944
#define BS 64
#define KVBLOCK 16
#define MAX_BLOCKS 128
#define GROUP 7                 // NH / NKV
#define ATTN_SCALE 0.08838834764831845f   // 128^-0.5
#define EPS 1e-6f
#define THETA 1000000.0f

typedef float v2f __attribute__((ext_vector_type(2)));
typedef float v8f __attribute__((ext_vector_type(8)));
typedef unsigned int v4u __attribute__((ext_vector_type(4)));
typedef int v4i __attribute__((ext_vector_type(4)));
typedef int v8i __attribute__((ext_vector_type(8)));

#if defined(__has_builtin)
#  if __has_builtin(__builtin_amdgcn_tensor_load_to_lds)
#    define HAVE_TDM 1
#  endif
#endif
#ifndef HAVE_TDM
#  define HAVE_TDM 0
#endif

// ---------------------------------------------------------------------------
// RMSNorm: out[b][:] = x[b][:] * rsqrt(mean(x^2)+eps) * w
// ---------------------------------------------------------------------------
__global__ __launch_bounds__(256) void rmsnorm_kernel(const float* __restrict__ x,
                                                      const float* __restrict__ w,
                                                      float* __restrict__ out) {
  int b = blockIdx.x;
  const float* row = x + (size_t)b * HID;
  __shared__ float red[256];
  float s = 0.f;
  for (int i = threadIdx.x; i < HID; i += 256) { float v = row[i]; s += v * v; }
  red[threadIdx.x] = s;
  __syncthreads();
  for (int off = 128; off > 0; off >>= 1) {
    if (threadIdx.x < off) red[threadIdx.x] += red[threadIdx.x + off];
    __syncthreads();
  }
  float inv = rsqrtf(red[0] / (float)HID + EPS);
  for (int i = threadIdx.x; i < HID; i += 256) out[(size_t)b * HID + i] = row[i] * inv * w[i];
}

// ---------------------------------------------------------------------------
// TDM helper: DMA a 2D fp32 tile (rows x cols) with row stride `stride_elems`
// from global memory into LDS at byte offset `lds_byte`. LDS layout gets
// 1 DWORD of padding every 32 DWORDs (pad_interval=4 -> 32 dw, pad_amount=0
// -> 1 dw), i.e. a [rows][33]-float layout that kills LDS bank conflicts for
// the WMMA operand reads. Descriptor per CDNA5 ISA ch.8 (D# groups 0/1).
// Toolchain here is the 6-arg clang-23 form:
//   (v4u g0, v8i g1, v4i g2, v4i g3, v8i extra, i32 cpol)
// ---------------------------------------------------------------------------
#if HAVE_TDM
__device__ __forceinline__ unsigned int lds_byte_off(const void* p) {
  // generic (flat) pointers into the shared aperture carry the LDS byte
  // offset in their low 32 bits (ISA 10.2 aperture mapping)
  return (unsigned int)(uintptr_t)p;
}

__device__ __forceinline__ void tdm_load_tile_f32(const float* gptr,
                                                  unsigned int lds_byte,
                                                  unsigned int rows,
                                                  unsigned int cols,
                                                  unsigned int stride_elems) {
  unsigned long long ga = (unsigned long long)(uintptr_t)gptr;
  v4u g0;
  g0.x = 0x1u;                                   // count=1, user descriptor
  g0.y = lds_byte;                               // lds_addr (bytes)
  g0.z = (unsigned int)(ga & 0xFFFFFFFFu);       // global_addr[31:0]
  g0.w = (unsigned int)((ga >> 32) & 0x1FFFFFFu) // global_addr[56:32]
         | (2u << 30);                           // type=2 ("image")
  v8i g1;
  unsigned long long s0 = (unsigned long long)stride_elems;
  g1[0] = (int)((2u << 16)       // data_size = 4 bytes
                | (1u << 20)     // pad_enable
                | (4u << 22));   // pad_interval code 4 = every 32 DWORDs, pad_amount code 0 = 1 DWORD
  g1[1] = (int)((cols & 0xFFFFu) << 16);                         // tensor_dim0[15:0] @bits63:48
  g1[2] = (int)(((cols >> 16) & 0xFFFFu) | ((rows & 0xFFFFu) << 16)); // tensor_dim0 hi / tensor_dim1 lo
  g1[3] = (int)(((rows >> 16) & 0xFFFFu) | ((cols & 0xFFFFu) << 16)); // tensor_dim1 hi / tile_dim0
  g1[4] = (int)(rows & 0xFFFFu);                                 // tile_dim1 (tile_dim2 = 0)
  g1[5] = (int)(unsigned int)(s0 & 0xFFFFFFFFu);                 // tensor_dim0_stride[31:0]
  g1[6] = (int)(unsigned int)((s0 >> 32) & 0xFFFFu);             // stride[47:32], dim1_stride lo = 0
  g1[7] = 0;
  v4i gz4 = {0, 0, 0, 0};
  v8i gz8 = {0, 0, 0, 0, 0, 0, 0, 0};
  __builtin_amdgcn_tensor_load_to_lds(g0, g1, gz4, gz4, gz8, 0);
}
#endif

// ---------------------------------------------------------------------------
// WMMA fp32 GEMM: out[64][N] = A[64][K] @ W[N][K]^T (+bias) (+residual)
// Block: 256 thr = 8 waves; block computes a 64x32 tile; wave -> 16x16 tile.
// TDM path: double-buffered LDS tiles filled by tensor_load_to_lds issued by
// wave 0 (TENSORcnt), overlapping the next chunk's DMA with current WMMAs.
// LDS row stride = KC+1 floats (matches TDM pad config) for bank-conflict-
// free operand reads. Per-lane WMMA operand packing per ISA 7.12.2.
// ---------------------------------------------------------------------------
#define KC 32
__global__ __launch_bounds__(256) void gemm_wmma_kernel(
    const float* __restrict__ A,     // [64][K]
    const float* __restrict__ W,     // [N][K] row-major
    const float* __restrict__ bias,  // [N] or nullptr
    const float* __restrict__ resid, // [64][N] or nullptr
    float* __restrict__ out,         // [64][N]
    int N, int K) {
  __shared__ __align__(16) float As[2][64][KC + 1];
  __shared__ __align__(16) float Bs[2][32][KC + 1];

  const int n0    = blockIdx.x * 32;
  const int tid   = threadIdx.x;
  const int wave  = tid >> 5;
  const int lane  = tid & 31;
  const int mtile = wave & 3;   // which 16 rows of the 64-row batch
  const int nsub  = wave >> 2;  // which 16 cols of the 32-col block tile
  const int lrow  = lane & 15;
  const int khalf = lane >> 4;  // lanes 16-31 carry K+2,K+3 per ISA layout

  v8f c = {0.f, 0.f, 0.f, 0.f, 0.f, 0.f, 0.f, 0.f};
  const int nch = K / KC;

#if HAVE_TDM
  // ---- prologue: DMA chunk 0 into buffer 0 ----
  if (wave == 0) {
    tdm_load_tile_f32(A, lds_byte_off(&As[0][0][0]), 64, KC, (unsigned)K);
    tdm_load_tile_f32(W + (size_t)n0 * K, lds_byte_off(&Bs[0][0][0]), 32, KC, (unsigned)K);
  }
  __builtin_amdgcn_s_wait_tensorcnt(0);
  __syncthreads();

  for (int i = 0; i < nch; ++i) {
    const int cur = i & 1;
    if ((wave == 0) && (i + 1 < nch)) {
      const int k0n = (i + 1) * KC;
      tdm_load_tile_f32(A + k0n, lds_byte_off(&As[cur ^ 1][0][0]), 64, KC, (unsigned)K);
      tdm_load_tile_f32(W + (size_t)n0 * K + k0n, lds_byte_off(&Bs[cur ^ 1][0][0]),
                        32, KC, (unsigned)K);
    }
#pragma unroll
    for (int kk = 0; kk < KC; kk += 4) {
      const int kb = kk + khalf * 2;
      v2f a, bm;
      a.x  = As[cur][mtile * 16 + lrow][kb];
      a.y  = As[cur][mtile * 16 + lrow][kb + 1];
      bm.x = Bs[cur][nsub * 16 + lrow][kb];      // B[k][n] = W[n][k]
      bm.y = Bs[cur][nsub * 16 + lrow][kb + 1];
      c = __builtin_amdgcn_wmma_f32_16x16x4_f32(false, a, false, bm,
                                                (short)0, c, false, false);
    }
    if (i + 1 < nch) {
      __builtin_amdgcn_s_wait_tensorcnt(0);  // wave0: next tiles landed (no-op for others)
      __syncthreads();
    }
  }
#else
  // ---- fallback: cooperative staging (round-1 path) ----
  for (int k0 = 0; k0 < K; k0 += KC) {
    for (int idx = tid; idx < 64 * KC; idx += 256) {
      int r = idx >> 5, col = idx & 31;
      As[0][r][col] = A[(size_t)r * K + k0 + col];
    }
    for (int idx = tid; idx < 32 * KC; idx += 256) {
      int r = idx >> 5, col = idx & 31;
      Bs[0][r][col] = W[(size_t)(n0 + r) * K + k0 + col];
    }
    __syncthreads();
#pragma unroll
    for (int kk = 0; kk < KC; kk += 4) {
      const int kb = kk + khalf * 2;
      v2f a, bm;
      a.x  = As[0][mtile * 16 + lrow][kb];
      a.y  = As[0][mtile * 16 + lrow][kb + 1];
      bm.x = Bs[0][nsub * 16 + lrow][kb];
      bm.y = Bs[0][nsub * 16 + lrow][kb + 1];
      c = __builtin_amdgcn_wmma_f32_16x16x4_f32(false, a, false, bm,
                                                (short)0, c, false, false);
    }
    __syncthreads();
  }
#endif

  // C/D layout: VGPR r -> (M = r + 8*khalf, N = lane&15)
#pragma unroll
  for (int r = 0; r < 8; r++) {
    int m = mtile * 16 + khalf * 8 + r;
    int n = n0 + nsub * 16 + lrow;
    float v = c[r];
    if (bias)  v += bias[n];
    if (resid) v += resid[(size_t)m * N + n];
    out[(size_t)m * N + n] = v;
  }
}

// ---------------------------------------------------------------------------
// RoPE (half-rotate, neox style) applied in place to q and k scratch.
// grid (BS, NH+NKV), block 64 (i = rotary pair index 0..63)
// ---------------------------------------------------------------------------
__global__ __launch_bounds__(64) void rope_kernel(float* __restrict__ q,
                                                  float* __restrict__ k,
                                                  const int* __restrict__ positions) {
  int b = blockIdx.x;
  int h = blockIdx.y;
  int i = threadIdx.x;  // 0..63
  float pos  = (float)positions[b];
  float freq = powf(THETA, -((float)i) / 64.0f);  // THETA^(-2i/HD)
  float ang  = pos * freq;
  float cs = cosf(ang), sn = sinf(ang);
  float* row = (h < NH) ? (q + ((size_t)b * NH + h) * HD)
                        : (k + ((size_t)b * NKV + (h - NH)) * HD);
  float t1 = row[i], t2 = row[i + 64];
  row[i]      = t1 * cs - t2 * sn;
  row[i + 64] = t2 * cs + t1 * sn;
}

// ---------------------------------------------------------------------------
// Paged GQA decode attention. grid (BS, NKV), block 256 (8 waves).
// Each wave walks positions p = wave, wave+8, ... with online softmax over
// GROUP=7 query heads; lane owns d = lane*4..lane*4+3. Position seq-1 reads
// the freshly computed k/v (inputs are never mutated).
// ---------------------------------------------------------------------------
__global__ __launch_bounds__(256) void attn_kernel(
    const float* __restrict__ q,     // [64][28][128] (roped)
    const float* __restrict__ knew,  // [64][4][128] (roped)
    const float* __restrict__ vnew,  // [64][4][128]
    const float* __restrict__ kc,    // [NUM_BLOCKS*16][4][128]
    const float* __restrict__ vc,
    const int* __restrict__ block_tables,  // [64][128]
    const int* __restrict__ seq_lens,      // [64]
    float* __restrict__ outp) {            // [64][28][128]
  const int b = blockIdx.x, kvh = blockIdx.y;
  const int tid = threadIdx.x, wave = tid >> 5, lane = tid & 31;

  __shared__ float qs[GROUP][HD];
  __shared__ float accs[8][GROUP][HD];
  __shared__ float ms[8], ls[8];

  for (int idx = tid; idx < GROUP * HD; idx += 256) {
    int h = idx / HD, d = idx % HD;
    qs[h][d] = q[((size_t)b * NH + kvh * GROUP + h) * HD + d];
  }
  __syncthreads();

  const int seq = seq_lens[b];
  float m = -1e30f, l = 0.f;
  float acc[GROUP][4];
#pragma unroll
  for (int h = 0; h < GROUP; h++)
#pragma unroll
    for (int j = 0; j < 4; j++) acc[h][j] = 0.f;

  const int d0 = lane * 4;

  for (int p = wave; p < seq; p += 8) {
    const float *kp, *vp;
    if (p == seq - 1) {
      kp = knew + ((size_t)b * NKV + kvh) * HD;
      vp = vnew + ((size_t)b * NKV + kvh) * HD;
    } else {
      int blk = block_tables[b * MAX_BLOCKS + (p >> 4)];
      size_t slot = (size_t)blk * KVBLOCK + (p & 15);
      kp = kc + (slot * NKV + kvh) * HD;
      vp = vc + (slot * NKV + kvh) * HD;
    }
    float4 kv = *(const float4*)(kp + d0);
    __builtin_prefetch(vp + d0, 0, 0);  // global_prefetch: overlap V fetch with reduction
    float s[GROUP];
#pragma unroll
    for (int h = 0; h < GROUP; h++) {
      float t = qs[h][d0] * kv.x + qs[h][d0 + 1] * kv.y +
                qs[h][d0 + 2] * kv.z + qs[h][d0 + 3] * kv.w;
#pragma unroll
      for (int off = 16; off > 0; off >>= 1) t += __shfl_xor(t, off, 32);
      s[h] = t * ATTN_SCALE;
    }
    float mx = m;
#pragma unroll
    for (int h = 0; h < GROUP; h++) mx = fmaxf(mx, s[h]);
    float corr = __expf(m - mx);
    float pr[GROUP];
    float psum = 0.f;
#pragma unroll
    for (int h = 0; h < GROUP; h++) { pr[h] = __expf(s[h] - mx); psum += pr[h]; }
    l = l * corr + psum;
    float4 vv = *(const float4*)(vp + d0);
#pragma unroll
    for (int h = 0; h < GROUP; h++) {
      acc[h][0] = acc[h][0] * corr + pr[h] * vv.x;
      acc[h][1] = acc[h][1] * corr + pr[h] * vv.y;
      acc[h][2] = acc[h][2] * corr + pr[h] * vv.z;
      acc[h][3] = acc[h][3] * corr + pr[h] * vv.w;
    }
    m = mx;
  }

  ms[wave] = m;
  ls[wave] = l;
#pragma unroll
  for (int h = 0; h < GROUP; h++)
#pragma unroll
    for (int j = 0; j < 4; j++) accs[wave][h][d0 + j] = acc[h][j];
  __syncthreads();

  // Cross-wave merge: 7 heads x 32 lanes = 224 active threads
  if (tid < GROUP * 32) {
    int h = tid >> 5, ln = tid & 31, dd = ln * 4;
    float M = -1e30f;
#pragma unroll
    for (int w = 0; w < 8; w++) M = fmaxf(M, ms[w]);
    float L = 0.f, o0 = 0.f, o1 = 0.f, o2 = 0.f, o3 = 0.f;
#pragma unroll
    for (int w = 0; w < 8; w++) {
      float e = __expf(ms[w] - M);
      L  += ls[w] * e;
      o0 += accs[w][h][dd]     * e;
      o1 += accs[w][h][dd + 1] * e;
      o2 += accs[w][h][dd + 2] * e;
      o3 += accs[w][h][dd + 3] * e;
    }
    float invL = 1.f / L;
    size_t base = ((size_t)b * NH + kvh * GROUP + h) * HD + dd;
    outp[base]     = o0 * invL;
    outp[base + 1] = o1 * invL;
    outp[base + 2] = o2 * invL;
    outp[base + 3] = o3 * invL;
  }
}

// ---------------------------------------------------------------------------
// silu(g) * u, written in place over g's buffer
// ---------------------------------------------------------------------------
__global__ __launch_bounds__(256) void silu_mul_kernel(float* __restrict__ g,
                                                       const float* __restrict__ u,
                                                       int n) {
  int i = blockIdx.x * 256 + threadIdx.x;
  if (i < n) {
    float gv = g[i];
    g[i] = (gv / (1.f + __expf(-gv))) * u[i];
  }
}

// ---------------------------------------------------------------------------
extern "C" void kernel_launch(void* const* d_in, const int* in_sizes, int n_in,
                              void* d_out, int out_size, void* d_ws, size_t ws_size,
                              hipStream_t stream) {
  const float* x            = (const float*)d_in[0];
  const int*   positions    = (const int*)d_in[1];
  const float* k_cache      = (const float*)d_in[2];
  const float* v_cache      = (const float*)d_in[3];
  const int*   block_tables = (const int*)d_in[4];
  const int*   seq_lens     = (const int*)d_in[5];
  // d_in[6] slot_mapping: implied by seq_lens-1, not needed
  const float* in_w   = (const float*)d_in[7];
  const float* post_w = (const float*)d_in[8];
  const float* Wq = (const float*)d_in[9];
  const float* bq = (const float*)d_in[10];
  const float* Wk = (const float*)d_in[11];
  const float* bk = (const float*)d_in[12];
  const float* Wv = (const float*)d_in[13];
  const float* bv = (const float*)d_in[14];
  const float* Wo = (const float*)d_in[15];
  const float* Wg = (const float*)d_in[16];
  const float* Wu = (const float*)d_in[17];
  const float* Wd = (const float*)d_in[18];
  float* out = (float*)d_out;

  float* ws   = (float*)d_ws;
  float* h    = ws; ws += (size_t)BS * HID;
  float* qb   = ws; ws += (size_t)BS * NH * HD;
  float* kb   = ws; ws += (size_t)BS * NKV * HD;
  float* vb   = ws; ws += (size_t)BS * NKV * HD;
  float* attn = ws; ws += (size_t)BS * NH * HD;
  float* hid  = ws; ws += (size_t)BS * HID;
  float* h2   = ws; ws += (size_t)BS * HID;
  float* g    = ws; ws += (size_t)BS * INTER;
  float* u    = ws; ws += (size_t)BS * INTER;

  // 1) input RMSNorm
  rmsnorm_kernel<<<BS, 256, 0, stream>>>(x, in_w, h);
  // 2) QKV projections (WMMA + TDM staging)
  gemm_wmma_kernel<<<(NH * HD) / 32, 256, 0, stream>>>(h, Wq, bq, nullptr, qb, NH * HD, HID);
  gemm_wmma_kernel<<<(NKV * HD) / 32, 256, 0, stream>>>(h, Wk, bk, nullptr, kb, NKV * HD, HID);
  gemm_wmma_kernel<<<(NKV * HD) / 32, 256, 0, stream>>>(h, Wv, bv, nullptr, vb, NKV * HD, HID);
  // 3) RoPE on q and k
  rope_kernel<<<dim3(BS, NH + NKV), 64, 0, stream>>>(qb, kb, positions);
  // 4) paged GQA attention (cache + fresh kv at seq-1)
  attn_kernel<<<dim3(BS, NKV), 256, 0, stream>>>(qb, kb, vb, k_cache, v_cache,
                                                 block_tables, seq_lens, attn);
  // 5) output projection + residual
  gemm_wmma_kernel<<<HID / 32, 256, 0, stream>>>(attn, Wo, nullptr, x, hid, HID, NH * HD);
  // 6) post-attention RMSNorm
  rmsnorm_kernel<<<BS, 256, 0, stream>>>(hid, post_w, h2);
  // 7) MLP gate / up (WMMA), silu*mul, down + residual
  gemm_wmma_kernel<<<INTER / 32, 256, 0, stream>>>(h2, Wg, nullptr, nullptr, g, INTER, HID);
  gemm_wmma_kernel<<<INTER / 32, 256, 0, stream>>>(h2, Wu, nullptr, nullptr, u, INTER, HID);
  int n = BS * INTER;
  silu_mul_kernel<<<(n + 255) / 256, 256, 0, stream>>>(g, u, n);
  gemm_wmma_kernel<<<HID / 32, 256, 0, stream>>>(g, Wd, nullptr, hid, out, HID, INTER);
}